// DistanceGreedyModel_75694503624834
// MI455X (gfx1250) — compile-verified
//
#include <hip/hip_runtime.h>

#define BIG 1.0e6f

// One workgroup per batch. Wave 0 (32 lanes) runs the sequential greedy chain;
// waves 1..7 warm L2 with the batch's 4MB distance slice via global_prefetch_b8.
// Assumes N <= 1024 (reference: N = 1000).
__global__ __launch_bounds__(256, 1) void greedy_nn_kernel(
    const float* __restrict__ distance,   // (B, N, N)
    const int*   __restrict__ mask,       // (B, N)  nonzero = visited/infeasible
    const int*   __restrict__ start_idx,  // (B,)
    const int*   __restrict__ pad_value,  // (1,)
    float*       __restrict__ out,        // pred (B,N) then pred_len (B,)
    int B, int N)
{
    const int b   = blockIdx.x;
    const int tid = threadIdx.x;
    const float* dmat = distance + (size_t)b * N * N;

    if (tid >= 32) {
        // ---- L2 warm path (7 waves): prefetch this batch's whole slice.
        // global_prefetch_b8: no VGPR return, no counter change, pulls full lines.
        const char*  base  = (const char*)dmat;
        const size_t bytes = (size_t)N * N * sizeof(float);
        for (size_t off = (size_t)(tid - 32) * 128; off < bytes; off += (size_t)224 * 128) {
            __builtin_prefetch(base + off, 0, 1);
        }
        return;
    }

    // ---- Sequential greedy chain: single wave32, no barriers.
    const int lane = tid;
    const int nb0  = lane * 32;          // this lane owns nodes [nb0, nb0+32)

    // Visited bitmask built with wave32 ballots: word w bit k = mask[32w + k].
    // Lane k supplies node 32w+k (coalesced load); nodes >= N forced visited.
    unsigned int m = 0u;
    for (int w = 0; w < 32; ++w) {
        const int node = (w << 5) + lane;
        const int mv   = (node < N) ? mask[(size_t)b * N + node] : 1;
        const unsigned int word = (unsigned int)__ballot(mv != 0);
        if (lane == w) m = word;
    }

    // rem = number of initially-unvisited nodes = number of active steps.
    int rem = __popc(~m);
#pragma unroll
    for (int off = 16; off > 0; off >>= 1) rem += __shfl_xor(rem, off, 32);

    int point = start_idx[b];
    const int pad = pad_value[0];
    float* pred = out + (size_t)b * N;

    const bool vec4ok = ((N & 3) == 0);
    const int steps = (rem < N) ? rem : N;

    for (int t = 0; t < steps; ++t) {
        const float* row = dmat + (size_t)point * N;

        // 4 independent (val, idx) min-chains for ILP across 8 x float4 chunks.
        // Indices ascend within a chain, so strict '<' keeps the lowest index.
        float bv[4];
        int   bi[4];
#pragma unroll
        for (int e = 0; e < 4; ++e) { bv[e] = 2.0f * BIG; bi[e] = 0; }

#pragma unroll
        for (int c = 0; c < 8; ++c) {
            const int s = nb0 + c * 4;
            float vv[4];
            if (vec4ok && (s + 4 <= N)) {
                const float4 v = *(const float4*)(row + s);   // global_load_b128
                vv[0] = v.x; vv[1] = v.y; vv[2] = v.z; vv[3] = v.w;
            } else {
                vv[0] = (s + 0 < N) ? row[s + 0] : BIG;
                vv[1] = (s + 1 < N) ? row[s + 1] : BIG;
                vv[2] = (s + 2 < N) ? row[s + 2] : BIG;
                vv[3] = (s + 3 < N) ? row[s + 3] : BIG;
            }
#pragma unroll
            for (int e = 0; e < 4; ++e) {
                const int   k   = c * 4 + e;
                const float val = ((m >> k) & 1u) ? BIG : vv[e];
                if (val < bv[e]) { bv[e] = val; bi[e] = nb0 + k; }
            }
        }

        // Combine the 4 chains with exact lowest-index tie-break.
        float v0 = bv[0]; int i0 = bi[0];
        if (bv[1] < v0 || (bv[1] == v0 && bi[1] < i0)) { v0 = bv[1]; i0 = bi[1]; }
        float v1 = bv[2]; int i1 = bi[2];
        if (bv[3] < v1 || (bv[3] == v1 && bi[3] < i1)) { v1 = bv[3]; i1 = bi[3]; }
        if (v1 < v0 || (v1 == v0 && i1 < i0)) { v0 = v1; i0 = i1; }

        // Wave32 float-min (5 shuffle hops), then recover the winner's index:
        // lowest lane achieving the min owns the lowest node block -> exact
        // jnp.argmin (first-occurrence) semantics.
        float wv = v0;
#pragma unroll
        for (int off = 16; off > 0; off >>= 1)
            wv = fminf(wv, __shfl_xor(wv, off, 32));

        const unsigned int ball = (unsigned int)__ballot(v0 == wv);
        const int src = __builtin_ctz(ball);
        const int idx = __shfl(i0, src, 32);

        if (lane == 0) pred[t] = (float)idx;
        if ((idx >> 5) == lane) m |= 1u << (idx & 31);   // mark visited (owner lane)
        point = idx;
    }

    // Pad the unused slots; write pred_len.
    for (int t = steps + lane; t < N; t += 32) pred[t] = (float)pad;
    if (lane == 0) out[(size_t)B * N + b] = (float)steps;
}

extern "C" void kernel_launch(void* const* d_in, const int* in_sizes, int n_in,
                              void* d_out, int out_size, void* d_ws, size_t ws_size,
                              hipStream_t stream) {
    const float* distance  = (const float*)d_in[0];
    const int*   mask      = (const int*)d_in[1];
    const int*   start_idx = (const int*)d_in[2];
    const int*   pad_value = (const int*)d_in[3];
    float* out = (float*)d_out;

    const int B = in_sizes[2];                        // start_idx has B elements
    const int N = (B > 0) ? (in_sizes[1] / B) : 0;    // mask has B*N elements

    if (B <= 0 || N <= 0) return;
    greedy_nn_kernel<<<B, 256, 0, stream>>>(distance, mask, start_idx, pad_value,
                                            out, B, N);
}